// MoBoAligner_86706799771885
// MI455X (gfx1250) — compile-verified
//
#include <hip/hip_runtime.h>
#include <math.h>

#define BDIM 2
#define IDIM 128
#define JDIM 512
#define CDIM 256
#define NINF (-__builtin_inff())

typedef float v2f __attribute__((ext_vector_type(2)));
typedef float v8f __attribute__((ext_vector_type(8)));

__device__ __forceinline__ float lse2(float a, float b) {
    float m = fmaxf(a, b);
    if (m == NINF) return NINF;
    return m + __logf(__expf(a - m) + __expf(b - m));
}

// Inclusive Hillis-Steele logsumexp scan over n elements resident in shared mem.
// Caller must have written sh[] and __syncthreads()'d before calling.
__device__ __forceinline__ void scan_lse(float* sh, int t, int n, bool forward) {
    for (int d = 1; d < n; d <<= 1) {
        float a = sh[t];
        float b = forward ? ((t >= d) ? sh[t - d] : NINF)
                          : ((t + d < n) ? sh[t + d] : NINF);
        __syncthreads();
        sh[t] = lse2(a, b);
        __syncthreads();
    }
}

// ---------------------------------------------------------------------------
// K1: energy[b,i,j] = (dot(text[b,i,:], mel[b,j,:]) / 256 + noise) / temp
// One wave per 16x16 (i,j) tile, K-loop of V_WMMA_F32_16X16X4_F32.
// f32 A 16x4 layout (ISA 7.12.2): lane = M in 0..15 (both halves), half
// selects K pair {0,1}/{2,3}; B 4x16 mirrored with lane = N.
// ---------------------------------------------------------------------------
__global__ void energy_wmma(const float* __restrict__ text,
                            const float* __restrict__ mel,
                            const float* __restrict__ noise,
                            const float* __restrict__ ratio,
                            float* __restrict__ E) {
    const int TJ = JDIM / 16, TI = IDIM / 16;
    int tile = blockIdx.x;                  // B * TI * TJ tiles
    int b  = tile / (TI * TJ);
    int r  = tile % (TI * TJ);
    int i0 = (r / TJ) * 16;
    int j0 = (r % TJ) * 16;
    int lane = threadIdx.x;                 // wave32
    int mn   = lane & 15;
    int kb   = (lane >> 4) * 2;

    const float* Arow = text + ((size_t)b * IDIM + i0 + mn) * CDIM;
    const float* Brow = mel  + ((size_t)b * JDIM + j0 + mn) * CDIM;

    v8f acc = {};
    for (int c0 = 0; c0 < CDIM; c0 += 4) {
        v2f a  = *(const v2f*)(Arow + c0 + kb);
        v2f bm = *(const v2f*)(Brow + c0 + kb);
        acc = __builtin_amdgcn_wmma_f32_16x16x4_f32(
            false, a, false, bm, (short)0, acc, false, false);
    }
    float invt = 1.0f / (0.1f + 0.9f * ratio[0]);   // TEMP_MIN..TEMP_MAX
    const float scale = 1.0f / 256.0f;              // 1/sqrt(C*C)
    int half = lane >> 4;
    int j = j0 + mn;
    for (int rr = 0; rr < 8; ++rr) {
        int i = i0 + half * 8 + rr;                 // C/D layout: M = r + half*8
        size_t idx = ((size_t)b * IDIM + i) * JDIM + j;
        E[idx] = (acc[rr] * scale + noise[idx]) * invt;
    }
}

// ---------------------------------------------------------------------------
// K2: per (b,i) row: P[k] = LSE_{j<=k} E[j]  (prefix), S[k] = LSE_{j>=k} E[j]
// ---------------------------------------------------------------------------
__global__ void rowscan(const float* __restrict__ E,
                        float* __restrict__ P, float* __restrict__ S) {
    __shared__ float sh[JDIM];
    int row = blockIdx.x;                   // b*I + i
    int j = threadIdx.x;
    float e = E[(size_t)row * JDIM + j];
    sh[j] = e; __syncthreads();
    scan_lse(sh, j, JDIM, true);
    P[(size_t)row * JDIM + j] = sh[j];
    __syncthreads();
    sh[j] = e; __syncthreads();
    scan_lse(sh, j, JDIM, false);
    S[(size_t)row * JDIM + j] = sh[j];
}

// ---------------------------------------------------------------------------
// K3: serial DP over i. dir 0 = alpha (forward), dir 1 = beta (backward).
// alpha: row[j] = LSE( E[i,j] + prefLSE_{k<=j}(carry[k]-S[i,k]),
//                      -10 + LSE_{k>j} carry[k] )            (invalid fill)
// beta:  row[j] = LSE( E[i,j] + sufLSE_{j<=k<=510}(carry[k+1]-P[i,k]),
//                      -10 + LSE_{1<=m<=j} carry[m] )
// ---------------------------------------------------------------------------
__global__ void dp_scan(const float* __restrict__ E,
                        const float* __restrict__ S,
                        const float* __restrict__ P,
                        float* __restrict__ alpha,
                        float* __restrict__ beta) {
    __shared__ float carry[JDIM];
    __shared__ float sh[JDIM];
    int b = blockIdx.x;
    int j = threadIdx.x;
    if (blockIdx.y == 0) {                          // ---- alpha ----
        carry[j] = (j == 0) ? 0.0f : NINF;
        __syncthreads();
        for (int i = 0; i < IDIM; ++i) {
            size_t base = ((size_t)b * IDIM + i) * JDIM + j;
            float Ev = E[base], Sv = S[base];
            sh[j] = carry[j] - Sv; __syncthreads();
            scan_lse(sh, j, JDIM, true);
            float pre = sh[j]; __syncthreads();
            sh[j] = carry[j]; __syncthreads();
            scan_lse(sh, j, JDIM, false);
            float sufx = (j < JDIM - 1) ? sh[j + 1] : NINF;  // exclusive
            __syncthreads();
            float row = lse2(Ev + pre, -10.0f + sufx);
            alpha[base] = row;
            sh[j] = row; __syncthreads();
            carry[j] = (j == 0) ? NINF : sh[j - 1];          // shift right
            __syncthreads();
        }
    } else {                                         // ---- beta ----
        carry[j] = (j == JDIM - 1) ? 0.0f : NINF;
        beta[((size_t)b * IDIM + (IDIM - 1)) * JDIM + j] = carry[j];
        __syncthreads();
        for (int i = IDIM - 2; i >= 0; --i) {
            size_t base = ((size_t)b * IDIM + i) * JDIM + j;
            float Ev = E[base], Pv = P[base];
            float tk = (j < JDIM - 1) ? (carry[j + 1] - Pv) : NINF;
            __syncthreads();
            sh[j] = tk; __syncthreads();
            scan_lse(sh, j, JDIM, false);
            float suf = sh[j]; __syncthreads();
            sh[j] = (j == 0) ? NINF : carry[j]; __syncthreads();
            scan_lse(sh, j, JDIM, true);
            float pre = sh[j]; __syncthreads();
            float row = lse2(Ev + suf, -10.0f + pre);
            beta[base] = row;
            __syncthreads();
            carry[j] = row;
            __syncthreads();
        }
    }
}

// ---------------------------------------------------------------------------
// K4: gamma = (alpha+beta) - LSE_i(alpha+beta), masked; expg = exp(gamma).
// One 128-thread block per (b,j) column.
// ---------------------------------------------------------------------------
__global__ void gamma_norm(const float* __restrict__ alpha,
                           const float* __restrict__ beta,
                           const float* __restrict__ text_mask,
                           const float* __restrict__ mel_mask,
                           float* __restrict__ gamma_out,
                           float* __restrict__ expg) {
    __shared__ float red[IDIM];
    int b = blockIdx.x / JDIM;
    int j = blockIdx.x % JDIM;
    int i = threadIdx.x;
    size_t idx = ((size_t)b * IDIM + i) * JDIM + j;
    bool ok = (text_mask[b * IDIM + i] * mel_mask[b * JDIM + j]) > 0.0f;
    float g = ok ? (alpha[idx] + beta[idx]) : NINF;
    red[i] = g; __syncthreads();
    for (int d = IDIM >> 1; d > 0; d >>= 1) {
        if (i < d) red[i] = fmaxf(red[i], red[i + d]);
        __syncthreads();
    }
    float m = red[0]; __syncthreads();
    red[i] = (m == NINF) ? 0.0f : __expf(g - m);
    __syncthreads();
    for (int d = IDIM >> 1; d > 0; d >>= 1) {
        if (i < d) red[i] += red[i + d];
        __syncthreads();
    }
    float lse = (m == NINF) ? NINF : (m + __logf(red[0]));
    float gm = (ok && lse != NINF) ? (g - lse) : NINF;
    gamma_out[idx] = gm;
    expg[idx] = (gm == NINF) ? 0.0f : __expf(gm);
}

// ---------------------------------------------------------------------------
// K5: expanded[b,j,c] = sum_i exp(gamma[b,i,j]) * text[b,i,c], * mel_mask.
// One wave per 16x16 (j,c) tile; K = I = 128 via f32 WMMA.
// ---------------------------------------------------------------------------
__global__ void expand_wmma(const float* __restrict__ expg,
                            const float* __restrict__ text,
                            const float* __restrict__ mel_mask,
                            float* __restrict__ out) {
    const int TC = CDIM / 16, TJ = JDIM / 16;
    int tile = blockIdx.x;                  // B * TJ * TC
    int b  = tile / (TJ * TC);
    int r  = tile % (TJ * TC);
    int j0 = (r / TC) * 16;
    int c0 = (r % TC) * 16;
    int lane = threadIdx.x;
    int mn   = lane & 15;
    int kb   = (lane >> 4) * 2;
    v8f acc = {};
    for (int k0 = 0; k0 < IDIM; k0 += 4) {
        int k = k0 + kb;
        v2f a, bm;
        a.x  = expg[((size_t)b * IDIM + k    ) * JDIM + j0 + mn];  // A = exp(gamma)^T
        a.y  = expg[((size_t)b * IDIM + k + 1) * JDIM + j0 + mn];
        bm.x = text[((size_t)b * IDIM + k    ) * CDIM + c0 + mn];
        bm.y = text[((size_t)b * IDIM + k + 1) * CDIM + c0 + mn];
        acc = __builtin_amdgcn_wmma_f32_16x16x4_f32(
            false, a, false, bm, (short)0, acc, false, false);
    }
    int half = lane >> 4;
    int c = c0 + mn;
    for (int rr = 0; rr < 8; ++rr) {
        int j = j0 + half * 8 + rr;
        float mm = mel_mask[b * JDIM + j];
        out[(size_t)BDIM * IDIM * JDIM + ((size_t)b * JDIM + j) * CDIM + c] =
            acc[rr] * mm;
    }
}

extern "C" void kernel_launch(void* const* d_in, const int* in_sizes, int n_in,
                              void* d_out, int out_size, void* d_ws, size_t ws_size,
                              hipStream_t stream) {
    const float* text  = (const float*)d_in[0];
    const float* mel   = (const float*)d_in[1];
    const float* tmask = (const float*)d_in[2];
    const float* mmask = (const float*)d_in[3];
    const float* noise = (const float*)d_in[4];
    const float* ratio = (const float*)d_in[5];
    float* out = (float*)d_out;
    float* ws  = (float*)d_ws;

    const size_t NE = (size_t)BDIM * IDIM * JDIM;   // 131072 floats
    float* E  = ws;            // energy
    float* P  = ws + 1 * NE;   // prefix LSE per (b,i)
    float* S  = ws + 2 * NE;   // suffix LSE per (b,i)
    float* A  = ws + 3 * NE;   // alpha
    float* Bt = ws + 4 * NE;   // beta
    float* G  = ws + 5 * NE;   // exp(gamma)

    energy_wmma<<<BDIM * (IDIM / 16) * (JDIM / 16), 32, 0, stream>>>(
        text, mel, noise, ratio, E);
    rowscan<<<BDIM * IDIM, JDIM, 0, stream>>>(E, P, S);
    dp_scan<<<dim3(BDIM, 2), JDIM, 0, stream>>>(E, S, P, A, Bt);
    gamma_norm<<<BDIM * JDIM, IDIM, 0, stream>>>(A, Bt, tmask, mmask, out, G);
    expand_wmma<<<BDIM * (JDIM / 16) * (CDIM / 16), 32, 0, stream>>>(
        G, text, mmask, out);
}